// SpatialSelectiveMRF_26010321945352
// MI455X (gfx1250) — compile-verified
//
#include <hip/hip_runtime.h>

typedef __attribute__((ext_vector_type(2))) float v2f;
typedef __attribute__((ext_vector_type(8))) float v8f;

#define LOG2PI   1.8378770664093453f
#define NS       60000
#define KCL      200
#define DZ       64
#define NNB      6
#define NTILES   13          // ceil(200/16)
#define CLIP_LO  1e-6f
#define CLIP_HI  (1.0f - 1e-6f)

// ---------------------------------------------------------------------------
// Kernel 1: per wave32, a 16-sample tile.
// U[s,k] = 0.5*(||x||^2 + ||mu||^2) - dot(x,mu) + 33.5*log(2pi)  (67 dims total)
// dot computed with V_WMMA_F32_16X16X4_F32 (16 chunks of dz + 1 spatial chunk).
// Then row softmax over K=200 (tiles padded to 208), expected energy -> out,
// clipped Q -> Qc workspace.
// ---------------------------------------------------------------------------
__global__ __launch_bounds__(32)
void mrf_point_kernel(const float* __restrict__ Z,
                      const float* __restrict__ S,
                      const float* __restrict__ sel_mu,
                      const float* __restrict__ spa_mu,
                      float* __restrict__ Qc,    // (NS, KCL)
                      float* __restrict__ out)   // (NS,) point energy
{
    const int lane = threadIdx.x;     // 0..31
    const int half = lane >> 4;       // 0|1 : selects K-pair within a chunk
    const int lrow = lane & 15;       // A row (M) / B column (N) index
    const int s0   = blockIdx.x * 16;

    // ---- A matrix (Z rows) per ISA 32-bit 16x4 layout: v0=K0|K2, v1=K1|K3 ----
    v2f a[16];
    float xsq = 0.f;
    const float* zrow = Z + (size_t)(s0 + lrow) * DZ;
#pragma unroll
    for (int kk = 0; kk < 16; ++kk) {
        a[kk] = *(const v2f*)(zrow + 4 * kk + 2 * half);
        xsq += a[kk].x * a[kk].x + a[kk].y * a[kk].y;
    }
    // spatial A chunk: lanes 0-15 hold (S0,S1), lanes 16-31 hold (S2,0)
    v2f as;
    const float* srow = S + (size_t)(s0 + lrow) * 3;
    as.x = srow[2 * half];
    as.y = half ? 0.f : srow[1];
    xsq += as.x * as.x + as.y * as.y;
    // combine lane halves -> full ||x||^2 of row lrow, present in all lanes
    xsq += __shfl_xor(xsq, 16, 32);

    // this thread's C-tile rows are m = r + 8*half; fetch ||x||^2 for each
    float xm[8];
#pragma unroll
    for (int r = 0; r < 8; ++r)
        xm[r] = __shfl(xsq, r + 8 * half, 32);

    float u[NTILES][8];   // fully unrolled -> stays in VGPRs

#pragma unroll
    for (int t = 0; t < NTILES; ++t) {
        const int  kcl = t * 16 + lrow;       // cluster handled by this lane (B col)
        const bool kv  = (kcl < KCL);
        const float* murow = sel_mu + (size_t)kcl * DZ;

        v8f c = {};
        float bsq = 0.f;
#pragma unroll
        for (int kk = 0; kk < 16; ++kk) {
            v2f b;
            if (kv) b = *(const v2f*)(murow + 4 * kk + 2 * half);
            else    { b.x = 0.f; b.y = 0.f; }
            bsq += b.x * b.x + b.y * b.y;
            c = __builtin_amdgcn_wmma_f32_16x16x4_f32(
                    false, a[kk], false, b, (short)0, c, false, false);
        }
        // spatial chunk
        v2f bs;
        const float* smrow = spa_mu + (size_t)kcl * 3;
        bs.x = kv ? smrow[2 * half] : 0.f;
        bs.y = (kv && !half) ? smrow[1] : 0.f;
        bsq += bs.x * bs.x + bs.y * bs.y;
        c = __builtin_amdgcn_wmma_f32_16x16x4_f32(
                false, as, false, bs, (short)0, c, false, false);

        // full ||mu||^2 for this lane's column (same column for all r)
        bsq += __shfl_xor(bsq, 16, 32);

#pragma unroll
        for (int r = 0; r < 8; ++r) {
            float uu = 0.5f * (xm[r] + bsq) - c[r] + 33.5f * LOG2PI;
            u[t][r] = kv ? uu : 1e30f;        // padded clusters -> Q = 0
        }
    }

    // ---- softmax over K (row = r + 8*half, columns spread over 16 lanes) ----
    float mx[8], sm[8], pt[8];
#pragma unroll
    for (int r = 0; r < 8; ++r) {
        float m = -u[0][r];
#pragma unroll
        for (int t = 1; t < NTILES; ++t) m = fmaxf(m, -u[t][r]);
#pragma unroll
        for (int d = 1; d < 16; d <<= 1) m = fmaxf(m, __shfl_xor(m, d, 32));
        mx[r] = m;
    }
#pragma unroll
    for (int r = 0; r < 8; ++r) {
        float s = 0.f;
#pragma unroll
        for (int t = 0; t < NTILES; ++t) s += expf(-u[t][r] - mx[r]);
#pragma unroll
        for (int d = 1; d < 16; d <<= 1) s += __shfl_xor(s, d, 32);
        sm[r] = s;
        pt[r] = 0.f;
    }

    // expected point energy + clipped Q store
#pragma unroll
    for (int t = 0; t < NTILES; ++t) {
        const int  kcl = t * 16 + lrow;
        const bool kv  = (kcl < KCL);
#pragma unroll
        for (int r = 0; r < 8; ++r) {
            float e = expf(-u[t][r] - mx[r]);
            float q = e / sm[r];
            pt[r] += q * u[t][r];             // padded: e==0 -> q==0 -> 0
            if (kv) {
                float qc = fminf(fmaxf(q, CLIP_LO), CLIP_HI);
                Qc[(size_t)(s0 + r + 8 * half) * KCL + kcl] = qc;
            }
        }
    }
#pragma unroll
    for (int r = 0; r < 8; ++r) {
#pragma unroll
        for (int d = 1; d < 16; d <<= 1) pt[r] += __shfl_xor(pt[r], d, 32);
        if (lrow == r)
            out[s0 + r + 8 * half] = pt[r];
    }
}

// ---------------------------------------------------------------------------
// Kernel 2: one wave32 per sample; lanes split K=200.
// co[j] = sum_k Qc[D[s,j],k] * Qc[s,k]  (Qc is 48MB -> L2 resident)
// out[s] += sum_j -log(co[j])
// ---------------------------------------------------------------------------
__global__ __launch_bounds__(256)
void mrf_doublet_kernel(const float* __restrict__ Qc,
                        const long long* __restrict__ D,
                        float* __restrict__ out)
{
    const int wave = threadIdx.x >> 5;
    const int lane = threadIdx.x & 31;
    const int sidx = blockIdx.x * 8 + wave;
    if (sidx >= NS) return;

    long long dj[NNB];
#pragma unroll
    for (int j = 0; j < NNB; ++j) {
        long long d = D[(size_t)sidx * NNB + j];
        dj[j] = d;
    }

    float co[NNB] = {0.f, 0.f, 0.f, 0.f, 0.f, 0.f};
    const float* qrow = Qc + (size_t)sidx * KCL;
    for (int k = lane; k < KCL; k += 32) {
        float qs = qrow[k];
#pragma unroll
        for (int j = 0; j < NNB; ++j) {
            long long idx = dj[j] < 0 ? 0 : dj[j];
            co[j] += Qc[(size_t)idx * KCL + k] * qs;
        }
    }
#pragma unroll
    for (int j = 0; j < NNB; ++j) {
#pragma unroll
        for (int d = 16; d >= 1; d >>= 1)
            co[j] += __shfl_xor(co[j], d, 32);
    }
    if (lane == 0) {
        float dbl = 0.f;
#pragma unroll
        for (int j = 0; j < NNB; ++j)
            if (dj[j] >= 0) dbl -= logf(co[j]);
        out[sidx] = out[sidx] + dbl;
    }
}

extern "C" void kernel_launch(void* const* d_in, const int* in_sizes, int n_in,
                              void* d_out, int out_size, void* d_ws, size_t ws_size,
                              hipStream_t stream) {
    const float*     Z      = (const float*)d_in[0];      // (60000,64)
    const float*     S      = (const float*)d_in[1];      // (60000,3)
    const long long* D      = (const long long*)d_in[2];  // (60000,6) int64
    const float*     sel_mu = (const float*)d_in[3];      // (200,64)
    const float*     spa_mu = (const float*)d_in[4];      // (200,3)
    float*           out    = (float*)d_out;              // (60000,)
    float*           Qc     = (float*)d_ws;               // 60000*200 floats = 48MB

    mrf_point_kernel<<<NS / 16, 32, 0, stream>>>(Z, S, sel_mu, spa_mu, Qc, out);
    mrf_doublet_kernel<<<(NS + 7) / 8, 256, 0, stream>>>(Qc, D, out);
}